// CollaborativeFeatureLearning_24567212933378
// MI455X (gfx1250) — compile-verified
//
#include <hip/hip_runtime.h>

// ---------------------------------------------------------------------------
// CollaborativeFeatureLearning fused CDNA5 implementation (gfx1250, wave32)
//
// Key identity: mean(bilinear_resize_4x(x)) == mean(x) exactly (half-pixel,
// integer scale, clamped edges -> every input pixel has total weight 16).
// Therefore the upsampled "modulation feature" tensor is never materialized;
// the transformer output is consumed only through a spatial mean.
//
// All GEMMs run on V_WMMA_F32_16X16X4_F32 (fp32-exact matrix path).
// ---------------------------------------------------------------------------

typedef __attribute__((ext_vector_type(2))) float v2f;
typedef __attribute__((ext_vector_type(8))) float v8f;

#define NPIX 9216      // 96*96 pixels per batch image

__device__ __forceinline__ v8f wmma4(v2f a, v2f b, v8f c) {
  // D(16x16,f32) += A(16x4,f32) * B(4x16,f32)
  return __builtin_amdgcn_wmma_f32_16x16x4_f32(
      /*neg_a=*/false, a, /*neg_b=*/false, b,
      /*c_mod=*/(short)0, c, /*reuse_a=*/false, /*reuse_b=*/false);
}

// One 16(d) x 16(pixel) tile of a 1x1 conv: W[64xC] * feat[C x 16].
// C is compile-time so the k-loop fully unrolls and all loads stay GLOBAL
// (no runtime pointer-array indexing -> no FLAT fallback).
template <int C>
__device__ __forceinline__ void align_tile(
    const float* __restrict__ w,      // [64][C] row-major
    const float* __restrict__ bias,   // [64]
    const float* __restrict__ feat,   // already offset to (b, c=0, s0)
    float* xs,                        // LDS [48][64]
    int e, int dt, int lo, int hi)
{
  const float* wrow = w + (size_t)(dt * 16 + lo) * C;   // A fragment: M = d
  const float* fcol = feat + lo;                        // B fragment: N = pixel
  v8f c = {};
  #pragma unroll
  for (int kk = 0; kk < C / 4; ++kk) {
    int kc = kk * 4 + 2 * hi;
    v2f a = *(const v2f*)(wrow + kc);
    v2f bfrag;
    bfrag.x = fcol[(size_t)kc * NPIX];
    bfrag.y = fcol[(size_t)(kc + 1) * NPIX];
    c = wmma4(a, bfrag, c);
  }
  #pragma unroll
  for (int r = 0; r < 8; ++r) {          // D: row M = r + 8*hi, col N = lo
    int d = dt * 16 + 8 * hi + r;
    xs[(lo * 3 + e) * 64 + d] = c[r] + bias[d];
  }
}

// ---------------------------------------------------------------------------
// Kernel: zero small accumulators in workspace
// ---------------------------------------------------------------------------
__global__ void cfl_zero(float* __restrict__ p, int n) {
  int i = blockIdx.x * blockDim.x + threadIdx.x;
  if (i < n) p[i] = 0.0f;
}

// ---------------------------------------------------------------------------
// Fused kernel: 1x1 align convs -> pre-norm MHA (E=3 tokens) -> pre-norm FFN
// -> per-(b,e,d) spatial partial sums (atomic) for later pooling.
// One workgroup = 128 threads (4 wave32 waves) = 16 pixels = 48 token rows.
// ---------------------------------------------------------------------------
__global__ __launch_bounds__(128)
void cfl_transformer(
    const float* __restrict__ feat_hat, const float* __restrict__ feat_dat,
    const float* __restrict__ feat_naf,
    const float* __restrict__ w_hat, const float* __restrict__ b_hat,
    const float* __restrict__ w_dat, const float* __restrict__ b_dat,
    const float* __restrict__ w_naf, const float* __restrict__ b_naf,
    const float* __restrict__ g1, const float* __restrict__ be1,
    const float* __restrict__ g2, const float* __restrict__ be2,
    const float* __restrict__ w_qkv, const float* __restrict__ b_qkv,
    const float* __restrict__ w_o,   const float* __restrict__ b_o,
    const float* __restrict__ w_f1,  const float* __restrict__ b_f1,
    const float* __restrict__ w_f2,  const float* __restrict__ b_f2,
    float* __restrict__ pooled)                 // [B][E][D] sums
{
  __shared__ float xs[48 * 64];    // token rows (residual stream)
  __shared__ float nb[48 * 64];    // LN output / attention output
  __shared__ float qb[48 * 192];   // qkv; reused as FFN hidden (48x128 halves)

  const int tid  = threadIdx.x;
  const int wv   = tid >> 5;        // wave id (wave32)
  const int lane = tid & 31;
  const int lo   = lane & 15;       // N / M-low index inside WMMA fragments
  const int hi   = lane >> 4;       // K-pair selector

  const int pix0 = blockIdx.x * 16;  // 9216 % 16 == 0 -> never straddles batch
  const int bb   = pix0 / NPIX;
  const int s0   = pix0 - bb * NPIX;

  // ---- Stage 1: 1x1 align convs; wave wv owns d-tile wv for all 3 experts -
  {
    const int dt = wv;   // 4 waves x 4 d-tiles; 106 k-steps per wave, balanced
    align_tile<180>(w_hat, b_hat, feat_hat + (size_t)bb * 180 * NPIX + s0,
                    xs, /*e=*/0, dt, lo, hi);
    align_tile<180>(w_dat, b_dat, feat_dat + (size_t)bb * 180 * NPIX + s0,
                    xs, /*e=*/1, dt, lo, hi);
    align_tile<64>(w_naf, b_naf, feat_naf + (size_t)bb * 64 * NPIX + s0,
                   xs, /*e=*/2, dt, lo, hi);
  }
  __syncthreads();

  // ---- Stage 2: LayerNorm 1 (per token row over D=64) ---------------------
  if (tid < 48) {
    int r = tid;
    float mu = 0.f;
    for (int d = 0; d < 64; ++d) mu += xs[r * 64 + d];
    mu *= (1.f / 64.f);
    float var = 0.f;
    for (int d = 0; d < 64; ++d) { float t = xs[r * 64 + d] - mu; var += t * t; }
    var *= (1.f / 64.f);
    float inv = rsqrtf(var + 1e-5f);
    for (int d = 0; d < 64; ++d)
      nb[r * 64 + d] = (xs[r * 64 + d] - mu) * inv * g1[d] + be1[d];
  }
  __syncthreads();

  // ---- Stage 3: QKV projection [48x64] x [64x192]^T -> qb -----------------
  for (int t = wv; t < 36; t += 4) {          // 3 row-tiles x 12 col-tiles
    int rt = t % 3, ct = t / 3;
    const float* arow = &nb[(rt * 16 + lo) * 64];
    const float* brow = w_qkv + (size_t)(ct * 16 + lo) * 64;   // [192][64]
    v8f c = {};
    #pragma unroll
    for (int kk = 0; kk < 16; ++kk) {
      int k = kk * 4 + 2 * hi;
      c = wmma4(*(const v2f*)(arow + k), *(const v2f*)(brow + k), c);
    }
    float bias = b_qkv[ct * 16 + lo];
    #pragma unroll
    for (int r = 0; r < 8; ++r)
      qb[(rt * 16 + 8 * hi + r) * 192 + ct * 16 + lo] = c[r] + bias;
  }
  __syncthreads();

  // ---- Stage 4: attention over E=3 tokens, 8 heads of dim 8 ---------------
  {
    int p = tid >> 3, h = tid & 7;            // 16 pixels x 8 heads = 128 thr
    float q[3][8], k[3][8], v[3][8];
    #pragma unroll
    for (int e = 0; e < 3; ++e) {
      const float* rowp = &qb[(p * 3 + e) * 192 + h * 8];
      #pragma unroll
      for (int i = 0; i < 8; ++i) {
        q[e][i] = rowp[i]; k[e][i] = rowp[64 + i]; v[e][i] = rowp[128 + i];
      }
    }
    const float scl = 0.35355339059327373f;   // 1/sqrt(8)
    #pragma unroll
    for (int s = 0; s < 3; ++s) {
      float s0s = 0.f, s1s = 0.f, s2s = 0.f;
      #pragma unroll
      for (int i = 0; i < 8; ++i) {
        s0s += q[s][i] * k[0][i]; s1s += q[s][i] * k[1][i]; s2s += q[s][i] * k[2][i];
      }
      s0s *= scl; s1s *= scl; s2s *= scl;
      float mx = fmaxf(s0s, fmaxf(s1s, s2s));
      float e0 = __expf(s0s - mx), e1 = __expf(s1s - mx), e2 = __expf(s2s - mx);
      float inv = 1.f / (e0 + e1 + e2);
      e0 *= inv; e1 *= inv; e2 *= inv;
      float* orow = &nb[(p * 3 + s) * 64 + h * 8];
      #pragma unroll
      for (int i = 0; i < 8; ++i)
        orow[i] = e0 * v[0][i] + e1 * v[1][i] + e2 * v[2][i];
    }
  }
  __syncthreads();

  // ---- Stage 5: output projection + residual into xs ----------------------
  for (int t = wv; t < 12; t += 4) {
    int rt = t % 3, ct = t / 3;
    const float* arow = &nb[(rt * 16 + lo) * 64];
    const float* brow = w_o + (size_t)(ct * 16 + lo) * 64;
    v8f c = {};
    #pragma unroll
    for (int kk = 0; kk < 16; ++kk) {
      int k = kk * 4 + 2 * hi;
      c = wmma4(*(const v2f*)(arow + k), *(const v2f*)(brow + k), c);
    }
    float bias = b_o[ct * 16 + lo];
    #pragma unroll
    for (int r = 0; r < 8; ++r) {
      int idx = (rt * 16 + 8 * hi + r) * 64 + ct * 16 + lo;
      xs[idx] += c[r] + bias;
    }
  }
  __syncthreads();

  // ---- Stage 6: LayerNorm 2 -----------------------------------------------
  if (tid < 48) {
    int r = tid;
    float mu = 0.f;
    for (int d = 0; d < 64; ++d) mu += xs[r * 64 + d];
    mu *= (1.f / 64.f);
    float var = 0.f;
    for (int d = 0; d < 64; ++d) { float t = xs[r * 64 + d] - mu; var += t * t; }
    var *= (1.f / 64.f);
    float inv = rsqrtf(var + 1e-5f);
    for (int d = 0; d < 64; ++d)
      nb[r * 64 + d] = (xs[r * 64 + d] - mu) * inv * g2[d] + be2[d];
  }
  __syncthreads();

  // ---- Stage 7: FFN (64 -> 256 GELU -> 64), hidden in two 128-wide halves -
  v8f cacc[3] = {};                           // FFN2 accumulators (3 tiles/wave)
  for (int half = 0; half < 2; ++half) {
    // FFN1 for hidden cols [half*128, half*128+128)
    for (int t = wv; t < 24; t += 4) {        // 3 row-tiles x 8 col-tiles
      int rt = t % 3, ct = t / 3;
      int col0 = half * 128 + ct * 16;
      const float* arow = &nb[(rt * 16 + lo) * 64];
      const float* brow = w_f1 + (size_t)(col0 + lo) * 64;     // [256][64]
      v8f c = {};
      #pragma unroll
      for (int kk = 0; kk < 16; ++kk) {
        int k = kk * 4 + 2 * hi;
        c = wmma4(*(const v2f*)(arow + k), *(const v2f*)(brow + k), c);
      }
      float bias = b_f1[col0 + lo];
      #pragma unroll
      for (int r = 0; r < 8; ++r) {
        float x = c[r] + bias;
        float g = 0.5f * x * (1.f + erff(x * 0.70710678118654752f)); // exact GELU
        qb[(rt * 16 + 8 * hi + r) * 128 + ct * 16 + lo] = g;
      }
    }
    __syncthreads();
    // FFN2 partial accumulation over this K half
    {
      int i = 0;
      for (int t = wv; t < 12; t += 4, ++i) {
        int rt = t % 3, ct = t / 3;
        const float* arow = &qb[(rt * 16 + lo) * 128];
        const float* brow = w_f2 + (size_t)(ct * 16 + lo) * 256 + half * 128; // [64][256]
        #pragma unroll
        for (int kk = 0; kk < 32; ++kk) {
          int k = kk * 4 + 2 * hi;
          cacc[i] = wmma4(*(const v2f*)(arow + k), *(const v2f*)(brow + k), cacc[i]);
        }
      }
    }
    __syncthreads();
  }
  // FFN2 bias + residual writeback
  {
    int i = 0;
    for (int t = wv; t < 12; t += 4, ++i) {
      int rt = t % 3, ct = t / 3;
      float bias = b_f2[ct * 16 + lo];
      #pragma unroll
      for (int r = 0; r < 8; ++r) {
        int idx = (rt * 16 + 8 * hi + r) * 64 + ct * 16 + lo;
        xs[idx] += cacc[i][r] + bias;
      }
    }
  }
  __syncthreads();

  // ---- Stage 8: spatial partial sums -> global pooled accumulators --------
  for (int i = tid; i < 192; i += 128) {      // E*D = 192 (e,d) pairs
    int e = i >> 6, d = i & 63;
    float s = 0.f;
    #pragma unroll
    for (int p = 0; p < 16; ++p) s += xs[(p * 3 + e) * 64 + d];
    atomicAdd(&pooled[(bb * 3 + e) * 64 + d], s);
  }
}

// ---------------------------------------------------------------------------
// Kernel: pooled sums -> sigmoid modulation -> per-(e,b,c) scale factors
// scale[e][b][j] = 1 + 0.2*(sigmoid(mean_pool . mod_w[e][j] + mod_b[e][j]) - 0.5)
// ---------------------------------------------------------------------------
__global__ void cfl_scales(const float* __restrict__ pooled,
                           const float* __restrict__ mod_w,
                           const float* __restrict__ mod_b,
                           float* __restrict__ scale) {
  int t = threadIdx.x;
  if (t >= 36) return;                        // B*E*3
  int b = t / 9, r = t % 9;
  int e = r / 3, j = r % 3;
  float acc = mod_b[e * 3 + j];
  const float* pw = mod_w + (size_t)(e * 3 + j) * 64;   // [E,3,D]
  const float* pp = pooled + (size_t)(b * 3 + e) * 64;
  for (int d = 0; d < 64; ++d)
    acc += pp[d] * (1.f / (float)NPIX) * pw[d];
  float m = 1.f / (1.f + __expf(-acc));
  scale[(e * 4 + b) * 3 + j] = 1.f + 0.2f * (m - 0.5f);
}

// ---------------------------------------------------------------------------
// Kernel: elementwise out * scale, clipped to [0,1]; float4 vectorized.
// One launch per expert e; 442368 float4 per launch (B*3*384*384 / 4).
// ---------------------------------------------------------------------------
__global__ __launch_bounds__(256)
void cfl_apply(const float4* __restrict__ src,
               const float* __restrict__ scale_e,   // 12 = B*3 channel scales
               float4* __restrict__ dst) {
  int i = blockIdx.x * blockDim.x + threadIdx.x;
  if (i >= 442368) return;
  int ch = i / 36864;                         // (b*3 + c); 147456 floats / channel
  float s = scale_e[ch];
  float4 v = src[i];
  v.x = fminf(fmaxf(v.x * s, 0.f), 1.f);
  v.y = fminf(fmaxf(v.y * s, 0.f), 1.f);
  v.z = fminf(fmaxf(v.z * s, 0.f), 1.f);
  v.w = fminf(fmaxf(v.w * s, 0.f), 1.f);
  dst[i] = v;
}

// ---------------------------------------------------------------------------
extern "C" void kernel_launch(void* const* d_in, const int* in_sizes, int n_in,
                              void* d_out, int out_size, void* d_ws, size_t ws_size,
                              hipStream_t stream) {
  (void)in_sizes; (void)n_in; (void)out_size; (void)ws_size;

  const float* feat_hat = (const float*)d_in[0];
  const float* feat_dat = (const float*)d_in[1];
  const float* feat_naf = (const float*)d_in[2];
  // d_in[3..5] = out0, out1, out2
  const float* w_hat = (const float*)d_in[6];
  const float* b_hat = (const float*)d_in[7];
  const float* w_dat = (const float*)d_in[8];
  const float* b_dat = (const float*)d_in[9];
  const float* w_naf = (const float*)d_in[10];
  const float* b_naf = (const float*)d_in[11];
  const float* g1  = (const float*)d_in[12];
  const float* be1 = (const float*)d_in[13];
  const float* g2  = (const float*)d_in[14];
  const float* be2 = (const float*)d_in[15];
  const float* w_qkv = (const float*)d_in[16];
  const float* b_qkv = (const float*)d_in[17];
  const float* w_o   = (const float*)d_in[18];
  const float* b_o   = (const float*)d_in[19];
  const float* w_f1  = (const float*)d_in[20];
  const float* b_f1  = (const float*)d_in[21];
  const float* w_f2  = (const float*)d_in[22];
  const float* b_f2  = (const float*)d_in[23];
  const float* mod_w = (const float*)d_in[24];
  const float* mod_b = (const float*)d_in[25];

  float* pooled = (float*)d_ws;           // 768 floats: [B][E][D] sums
  float* scale  = pooled + 768;           // 36 floats:  [E][B][3]

  cfl_zero<<<4, 256, 0, stream>>>(pooled, 768 + 36);

  // 36864 pixels / 16 per workgroup
  cfl_transformer<<<2304, 128, 0, stream>>>(
      feat_hat, feat_dat, feat_naf,
      w_hat, b_hat, w_dat, b_dat, w_naf, b_naf,
      g1, be1, g2, be2,
      w_qkv, b_qkv, w_o, b_o, w_f1, b_f1, w_f2, b_f2,
      pooled);

  cfl_scales<<<1, 64, 0, stream>>>(pooled, mod_w, mod_b, scale);

  for (int e = 0; e < 3; ++e) {
    const float4* src = (const float4*)d_in[3 + e];
    float4* dst = ((float4*)d_out) + (size_t)e * 442368;
    cfl_apply<<<1728, 256, 0, stream>>>(src, scale + e * 12, dst);
  }
}